// StochasticTwoLayerRGAT_35648228557576
// MI455X (gfx1250) — compile-verified
//
#include <hip/hip_runtime.h>
#include <hip/hip_bf16.h>

// ---------------------------------------------------------------------------
// Two-layer RGAT for MI455X (gfx1250, wave32).
//   per relation: feat = h @ W[r]   (WMMA f32 16x16x4, fp32 accumulate,
//                                    A-tile staged via GLOBAL_LOAD_ASYNC_TO_LDS)
//                 el/er = <feat, al/ar>          (wave-per-node reduction)
//                 edge softmax: 3 passes, float4-vectorized gathers, native
//                 f32 atomics (global_atomic_max_num_f32 / global_atomic_add_f32)
//   final:        out = h2 @ Wl + bl             (WMMA f32 16x16x4)
// Working set (~90MB) fits in the 192MB L2 -> scatter atomics are L2-resident.
// ---------------------------------------------------------------------------

#define NNODES 50000
#define NREL   3
#define NEDGE  300000
#define NHEAD  4
#define HDIM   32
#define FEATW  128   // NHEAD*HDIM == IN == HID
#define OUTW   64

typedef __attribute__((ext_vector_type(2))) float v2f;
typedef __attribute__((ext_vector_type(8))) float v8f;

// -------------------------- native f32 atomics -----------------------------
__device__ __forceinline__ void atomic_add_f32(float* addr, float val) {
  asm volatile("global_atomic_add_f32 %0, %1, off"
               :: "v"(addr), "v"(val) : "memory");
}
__device__ __forceinline__ void atomic_max_f32(float* addr, float val) {
  asm volatile("global_atomic_max_num_f32 %0, %1, off"
               :: "v"(addr), "v"(val) : "memory");
}

// ---------------- async global -> LDS copy (CDNA5, ASYNCcnt) ---------------
// LDS byte offset = low 32 bits of the generic pointer (AMDGPU generic
// pointers to LDS are {aperture_hi32, lds_offset32}).
__device__ __forceinline__ void async_copy_b128(unsigned lds_off,
                                                const float* gsrc) {
  asm volatile("global_load_async_to_lds_b128 %0, %1, off"
               :: "v"(lds_off), "v"(gsrc) : "memory");
}
__device__ __forceinline__ void wait_async0() {
  asm volatile("s_wait_asynccnt 0x0" ::: "memory");
}

// ------------------------------ fill ---------------------------------------
__global__ void fill_f32(float* __restrict__ p, float v, int n) {
  int i = blockIdx.x * blockDim.x + threadIdx.x;
  if (i < n) p[i] = v;
}

// ------------------------------ GEMM ---------------------------------------
// C[nrows, NCOL] = A[nrows, 128] * B[128, NCOL] (+ bias)
// One block = 16-row strip. NCOL/16 waves; wave w owns columns [16w, 16w+16).
// A-tile copied straight into padded LDS with async b128 copies (no VGPR
// round-trip), then consumed as WMMA fragments.
template <int NCOL>
__global__ void gemm_wmma_f32(const float* __restrict__ A,
                              const float* __restrict__ B,
                              float* __restrict__ C,
                              const float* __restrict__ bias) {
  constexpr int K   = FEATW;        // 128
  constexpr int NW  = NCOL / 16;    // waves per block
  constexpr int LDA = K + 4;        // 132: LDS pad kills 16-way bank conflict
  __shared__ float As[16 * LDA];

  const int tid  = threadIdx.x;     // 32*NW threads
  const int wave = tid >> 5;
  const int lane = tid & 31;
  const int m0   = blockIdx.x * 16; // 50000 = 3125*16, no tail

  // async-stage A tile: 16 rows x 128 cols = 512 chunks of 16B
  const unsigned lds_base = (unsigned)(uintptr_t)(&As[0]);
  for (int chunk = tid; chunk < 16 * (K / 4); chunk += 32 * NW) {
    int r  = chunk >> 5;            // 32 x 16B chunks per row
    int c4 = chunk & 31;
    async_copy_b128(lds_base + (unsigned)((r * LDA + c4 * 4) * 4),
                    A + (size_t)(m0 + r) * K + c4 * 4);
  }
  wait_async0();
  __syncthreads();

  const int n0    = wave * 16;
  const int row   = lane & 15;       // A: M index / B: N index
  const int khalf = (lane >> 4) * 2; // lanes 16-31 hold K+2,K+3

  v8f c = {};
  #pragma unroll 8
  for (int kb = 0; kb < K; kb += 4) {
    v2f a, b;
    a.x = As[row * LDA + kb + khalf + 0];
    a.y = As[row * LDA + kb + khalf + 1];
    b.x = B[(size_t)(kb + khalf + 0) * NCOL + n0 + row];
    b.y = B[(size_t)(kb + khalf + 1) * NCOL + n0 + row];
    c = __builtin_amdgcn_wmma_f32_16x16x4_f32(
        /*neg_a=*/false, a, /*neg_b=*/false, b,
        /*c_mod=*/(short)0, c, /*reuse_a=*/false, /*reuse_b=*/false);
  }

  // C/D layout: VGPR v -> M = v (lanes 0-15) or v+8 (lanes 16-31), N = lane%16
  const int cm = (lane >> 4) * 8;
  const int cn = lane & 15;
  float badd = bias ? bias[n0 + cn] : 0.0f;
  #pragma unroll
  for (int v = 0; v < 8; v++) {
    C[(size_t)(m0 + cm + v) * NCOL + n0 + cn] = c[v] + badd;
  }
}

// -------------------- attention coefficients (el/er) -----------------------
// one wave per node; lanes over D, loop over heads; shuffle reduction
__global__ void attn_coeff(const float* __restrict__ feat,
                           const float* __restrict__ al,
                           const float* __restrict__ ar,
                           float* __restrict__ el, float* __restrict__ er) {
  int node = blockIdx.x * (blockDim.x >> 5) + (threadIdx.x >> 5);
  int lane = threadIdx.x & 31;
  if (node >= NNODES) return;
  const float* f = feat + (size_t)node * FEATW;
  #pragma unroll
  for (int h = 0; h < NHEAD; h++) {
    float x  = f[h * HDIM + lane];
    float sl = x * al[h * HDIM + lane];
    float sr = x * ar[h * HDIM + lane];
    #pragma unroll
    for (int off = 16; off > 0; off >>= 1) {
      sl += __shfl_xor(sl, off, 32);
      sr += __shfl_xor(sr, off, 32);
    }
    if (lane == 0) {
      el[node * NHEAD + h] = sl;
      er[node * NHEAD + h] = sr;
    }
  }
}

// ------------------- edge pass 1: logits + segment max ---------------------
// float4-vectorized: one b64 (src,dst) + two b128 gathers + one b128 store
__global__ void edge_logits(const long long* __restrict__ src,
                            const long long* __restrict__ dst,
                            const float4* __restrict__ el4,
                            const float4* __restrict__ er4,
                            float4* __restrict__ ebuf4,
                            float* __restrict__ emax) {
  int e = blockIdx.x * blockDim.x + threadIdx.x;
  if (e >= NEDGE) return;
  long long s = src[e], d = dst[e];
  float4 l = el4[s];
  float4 r = er4[d];
  float4 v;
  v.x = l.x + r.x; v.x = (v.x > 0.0f) ? v.x : v.x * 0.2f;
  v.y = l.y + r.y; v.y = (v.y > 0.0f) ? v.y : v.y * 0.2f;
  v.z = l.z + r.z; v.z = (v.z > 0.0f) ? v.z : v.z * 0.2f;
  v.w = l.w + r.w; v.w = (v.w > 0.0f) ? v.w : v.w * 0.2f;
  ebuf4[e] = v;
  float* base = emax + d * NHEAD;
  atomic_max_f32(base + 0, v.x);
  atomic_max_f32(base + 1, v.y);
  atomic_max_f32(base + 2, v.z);
  atomic_max_f32(base + 3, v.w);
}

// ------------------- edge pass 2: exp + segment sum ------------------------
__global__ void edge_exp(const long long* __restrict__ dst,
                         const float4* __restrict__ emax4,
                         float4* __restrict__ ebuf4,
                         float* __restrict__ denom) {
  int e = blockIdx.x * blockDim.x + threadIdx.x;
  if (e >= NEDGE) return;
  long long d = dst[e];
  float4 v = ebuf4[e];
  float4 m = emax4[d];
  float4 ex;
  ex.x = __expf(v.x - m.x);
  ex.y = __expf(v.y - m.y);
  ex.z = __expf(v.z - m.z);
  ex.w = __expf(v.w - m.w);
  ebuf4[e] = ex;
  float* base = denom + d * NHEAD;
  atomic_add_f32(base + 0, ex.x);
  atomic_add_f32(base + 1, ex.y);
  atomic_add_f32(base + 2, ex.z);
  atomic_add_f32(base + 3, ex.w);
}

// ------------------- edge pass 3: weighted scatter -------------------------
// one wave per edge; lanes = D; loop over heads (coalesced 128B per head)
__global__ void edge_msg(const long long* __restrict__ src,
                         const long long* __restrict__ dst,
                         const float* __restrict__ feat,
                         const float4* __restrict__ ebuf4,
                         const float4* __restrict__ denom4,
                         float* __restrict__ acc) {
  int e    = blockIdx.x * (blockDim.x >> 5) + (threadIdx.x >> 5);
  int lane = threadIdx.x & 31;
  if (e >= NEDGE) return;
  long long s = src[e], d = dst[e];
  float4 ex = ebuf4[e];
  float4 dn = denom4[d];
  float alpha[NHEAD] = {ex.x / dn.x, ex.y / dn.y, ex.z / dn.z, ex.w / dn.w};
  #pragma unroll
  for (int h = 0; h < NHEAD; h++) {
    float val = feat[s * FEATW + h * HDIM + lane] * alpha[h];
    atomic_add_f32(acc + d * FEATW + h * HDIM + lane, val);
  }
}

// ------------------- layer finalize: + sum_r b[r], opt relu ----------------
__global__ void finalize(float* __restrict__ acc, const float* __restrict__ b,
                         int do_relu) {
  int i = blockIdx.x * blockDim.x + threadIdx.x;
  if (i >= NNODES * FEATW) return;
  int c = i & (FEATW - 1);
  float v = acc[i] + b[c] + b[FEATW + c] + b[2 * FEATW + c];
  if (do_relu) v = fmaxf(v, 0.0f);
  acc[i] = v;
}

// ---------------------------------------------------------------------------
static void rgat_layer(const float* hin, const long long* edges,
                       const float* W, const float* al, const float* ar,
                       const float* b, float* feat, float* el, float* er,
                       float* emax, float* denom, float* ebuf, float* acc,
                       int do_relu, hipStream_t stream) {
  const int NF = NNODES * FEATW;
  fill_f32<<<(NF + 255) / 256, 256, 0, stream>>>(acc, 0.0f, NF);

  for (int r = 0; r < NREL; r++) {
    const long long* srcp = edges + (size_t)r * 2 * NEDGE;
    const long long* dstp = srcp + NEDGE;

    gemm_wmma_f32<FEATW><<<NNODES / 16, 256, 0, stream>>>(
        hin, W + (size_t)r * FEATW * FEATW, feat, nullptr);

    attn_coeff<<<(NNODES + 7) / 8, 256, 0, stream>>>(
        feat, al + r * FEATW, ar + r * FEATW, el, er);

    const int NH = NNODES * NHEAD;
    fill_f32<<<(NH + 255) / 256, 256, 0, stream>>>(emax, -INFINITY, NH);
    fill_f32<<<(NH + 255) / 256, 256, 0, stream>>>(denom, 0.0f, NH);

    edge_logits<<<(NEDGE + 255) / 256, 256, 0, stream>>>(
        srcp, dstp, (const float4*)el, (const float4*)er,
        (float4*)ebuf, emax);
    edge_exp<<<(NEDGE + 255) / 256, 256, 0, stream>>>(
        dstp, (const float4*)emax, (float4*)ebuf, denom);
    edge_msg<<<(NEDGE + 7) / 8, 256, 0, stream>>>(
        srcp, dstp, feat, (const float4*)ebuf, (const float4*)denom, acc);
  }
  finalize<<<(NF + 255) / 256, 256, 0, stream>>>(acc, b, do_relu);
}

extern "C" void kernel_launch(void* const* d_in, const int* in_sizes, int n_in,
                              void* d_out, int out_size, void* d_ws,
                              size_t ws_size, hipStream_t stream) {
  const float*     x     = (const float*)d_in[0];
  const long long* edges = (const long long*)d_in[1];
  const float*     W1    = (const float*)d_in[2];
  const float*     al1   = (const float*)d_in[3];
  const float*     ar1   = (const float*)d_in[4];
  const float*     b1    = (const float*)d_in[5];
  const float*     W2    = (const float*)d_in[6];
  const float*     al2   = (const float*)d_in[7];
  const float*     ar2   = (const float*)d_in[8];
  const float*     b2    = (const float*)d_in[9];
  const float*     Wl    = (const float*)d_in[10];
  const float*     bl    = (const float*)d_in[11];
  float*           out   = (float*)d_out;

  // workspace layout (floats): ~85 MB total; all segments 16B-aligned
  float* ws    = (float*)d_ws;
  float* feat  = ws;                              // N*128
  float* acc1  = feat  + (size_t)NNODES * FEATW;  // N*128
  float* acc2  = acc1  + (size_t)NNODES * FEATW;  // N*128
  float* el    = acc2  + (size_t)NNODES * FEATW;  // N*4
  float* er    = el    + (size_t)NNODES * NHEAD;
  float* emax  = er    + (size_t)NNODES * NHEAD;
  float* denom = emax  + (size_t)NNODES * NHEAD;
  float* ebuf  = denom + (size_t)NNODES * NHEAD;  // E*4

  // layer 1: x -> acc1 (relu)
  rgat_layer(x, edges, W1, al1, ar1, b1, feat, el, er, emax, denom, ebuf,
             acc1, /*relu=*/1, stream);
  // layer 2: acc1 -> acc2 (no relu)
  rgat_layer(acc1, edges, W2, al2, ar2, b2, feat, el, er, emax, denom, ebuf,
             acc2, /*relu=*/0, stream);
  // final linear: out = acc2 @ Wl + bl   [N, 64]
  gemm_wmma_f32<OUTW><<<NNODES / 16, 128, 0, stream>>>(acc2, Wl, out, bl);
}